// RoiPoolingConv_53558242181890
// MI455X (gfx1250) — compile-verified
//
#include <hip/hip_runtime.h>
#include <math.h>
#include <stdint.h>

typedef __attribute__((ext_vector_type(4))) float v4f;

#define POOL 7
#define NPIX (POOL * POOL)
#define NUM_ROIS 512
#define IMG_H 64
#define IMG_W 64
#define IMG_C 1024

struct Coord {
    int   i0;
    int   i1;
    float f;
    int   pad;   // pad to 16B -> single b128 load
};

// ---------------------------------------------------------------------------
// CDNA5 async copy: one 16B element per lane, global -> LDS, tracked by
// ASYNCcnt (CDNA5 ISA). GV addressing mode:
//   global_load_async_to_lds_b128  vdst(lds addr), vaddr(64b), off
// ---------------------------------------------------------------------------
__device__ __forceinline__ void async_g2l_b128(uint32_t lds_byte_addr,
                                               const float* gptr) {
    uint64_t ga = (uint64_t)(uintptr_t)gptr;
    asm volatile("global_load_async_to_lds_b128 %0, %1, off"
                 :: "v"(lds_byte_addr), "v"(ga)
                 : "memory");
}

// ---------------------------------------------------------------------------
// Main kernel (defined first so the disasm snippet shows the CDNA5 async ops).
//
// One 256-thread block per ROI; loop over its 49 output pixels with a 3-buffer
// async pipeline keeping 2 pixels (8 async b128 ops) in flight. Each lane owns
// 4 channels (float4): taps for pixels p+1/p+2 stream into LDS via ASYNCcnt
// while pixel p is interpolated from its buffer. Lanes read back only their
// own 16B slices, so the only sync needed is s_wait_asynccnt (no barrier).
// The WAR hazard on buffer reuse ((p+2)%3 == (p-1)%3) is covered because
// iteration p-1's ds_load results are consumed (DScnt waited) before its NT
// store, which precedes iteration p's volatile async issues in program order.
//
// Output uses non-temporal b128 stores so the 98MB output stream does not
// evict the 16MB L2-resident image (192MB L2, 23.3 TB/s HBM write floor).
// ---------------------------------------------------------------------------
__global__ __launch_bounds__(256)
void roi_bilinear_async_kernel(const float* __restrict__ img,
                               const Coord* __restrict__ wsY,
                               const Coord* __restrict__ wsX,
                               float* __restrict__ out) {
    __shared__ v4f smem[3][4][256];   // 48 KB: [buffer][tap][lane]

    const int roi  = blockIdx.x;
    const int tid  = threadIdx.x;
    const int c    = tid * 4;         // 256 lanes * float4 = 1024 channels
    const int roiP = roi * POOL;

    // Low 32 bits of the generic LDS pointer == wave-relative LDS byte offset
    // (aperture bits live in the high dword per the flat-address mapping).
    const uint32_t lds_lane = (uint32_t)(uintptr_t)&smem[0][0][tid];
    const uint32_t TAP_STRIDE = 256 * 16;        // 4 KB per tap
    const uint32_t BUF_STRIDE = 4 * TAP_STRIDE;  // 16 KB per buffer

    const float* imgc = img + c;

    // Issue the 4 async taps for pixel p into buffer (p % 3).
    auto issue_taps = [&](int p) {
        int py = p / POOL;
        int px = p - py * POOL;
        Coord cy = wsY[roiP + py];
        Coord cx = wsX[roiP + px];
        uint32_t l = lds_lane + (uint32_t)(p % 3) * BUF_STRIDE;
        const float* r0 = imgc + (size_t)cy.i0 * (IMG_W * IMG_C);
        const float* r1 = imgc + (size_t)cy.i1 * (IMG_W * IMG_C);
        size_t o0 = (size_t)cx.i0 * IMG_C;
        size_t o1 = (size_t)cx.i1 * IMG_C;
        async_g2l_b128(l + 0 * TAP_STRIDE, r0 + o0);   // v00
        async_g2l_b128(l + 1 * TAP_STRIDE, r0 + o1);   // v01
        async_g2l_b128(l + 2 * TAP_STRIDE, r1 + o0);   // v10
        async_g2l_b128(l + 3 * TAP_STRIDE, r1 + o1);   // v11
    };

    // Prologue: 2 pixels in flight.
    issue_taps(0);
    issue_taps(1);

    for (int p = 0; p < NPIX; ++p) {
        if (p + 2 < NPIX) {
            issue_taps(p + 2);
            // 12 outstanding; keep the newest 8 (pixels p+1, p+2) in flight
            // => pixel p's 4 taps have landed in LDS.
            asm volatile("s_wait_asynccnt 0x8" ::: "memory");
        } else if (p + 1 < NPIX) {
            // 8 outstanding; keep the newest 4 (pixel p+1) in flight.
            asm volatile("s_wait_asynccnt 0x4" ::: "memory");
        } else {
            asm volatile("s_wait_asynccnt 0x0" ::: "memory");
        }

        const int buf = p % 3;
        v4f v00 = smem[buf][0][tid];
        v4f v01 = smem[buf][1][tid];
        v4f v10 = smem[buf][2][tid];
        v4f v11 = smem[buf][3][tid];

        int py = p / POOL;
        int px = p - py * POOL;
        float fy = wsY[roiP + py].f;
        float fx = wsX[roiP + px].f;
        v4f fxv = {fx, fx, fx, fx};
        v4f fyv = {fy, fy, fy, fy};

        v4f top = v00 + (v01 - v00) * fxv;
        v4f bot = v10 + (v11 - v10) * fxv;
        v4f res = top + (bot - top) * fyv;

        __builtin_nontemporal_store(
            res, (v4f*)(out + ((size_t)roi * NPIX + p) * IMG_C + c));
    }
}

// ---------------------------------------------------------------------------
// Stage 1: per-(roi, pool index) bilinear coordinates for both axes.
// ---------------------------------------------------------------------------
__global__ void roi_coords_kernel(const int* __restrict__ rois,
                                  Coord* __restrict__ wsY,
                                  Coord* __restrict__ wsX) {
    int t = blockIdx.x * blockDim.x + threadIdx.x;
    if (t >= NUM_ROIS * POOL) return;
    int roi = t / POOL;
    int p   = t - roi * POOL;

    int x1 = rois[roi * 4 + 0];
    int y1 = rois[roi * 4 + 1];
    int w  = rois[roi * 4 + 2];
    int h  = rois[roi * 4 + 3];
    int we = min(x1 + w, IMG_W) - x1;
    int he = min(y1 + h, IMG_H) - y1;

    {   // y axis
        float eff = (float)he;
        float tt  = (p + 0.5f) * eff / (float)POOL - 0.5f;
        float f0  = floorf(tt);
        Coord cc;
        cc.f   = tt - f0;  // frac uses unclipped f0, matching the reference
        cc.i0  = (int)fminf(fmaxf(f0, 0.0f), eff - 1.0f) + y1;
        cc.i1  = (int)fminf(fmaxf(f0 + 1.0f, 0.0f), eff - 1.0f) + y1;
        cc.pad = 0;
        wsY[t] = cc;
    }
    {   // x axis
        float eff = (float)we;
        float tt  = (p + 0.5f) * eff / (float)POOL - 0.5f;
        float f0  = floorf(tt);
        Coord cc;
        cc.f   = tt - f0;
        cc.i0  = (int)fminf(fmaxf(f0, 0.0f), eff - 1.0f) + x1;
        cc.i1  = (int)fminf(fmaxf(f0 + 1.0f, 0.0f), eff - 1.0f) + x1;
        cc.pad = 0;
        wsX[t] = cc;
    }
}

extern "C" void kernel_launch(void* const* d_in, const int* in_sizes, int n_in,
                              void* d_out, int out_size, void* d_ws, size_t ws_size,
                              hipStream_t stream) {
    const float* img  = (const float*)d_in[0];   // (1,64,64,1024) f32
    const int*   rois = (const int*)d_in[1];     // (512,4) i32
    float*       out  = (float*)d_out;           // (1,512,7,7,1024) f32

    Coord* wsY = (Coord*)d_ws;                   // 512*7 Coords
    Coord* wsX = wsY + NUM_ROIS * POOL;          // 512*7 Coords (~114KB total)

    int nCoord = NUM_ROIS * POOL;
    roi_coords_kernel<<<(nCoord + 255) / 256, 256, 0, stream>>>(rois, wsY, wsX);

    roi_bilinear_async_kernel<<<NUM_ROIS, 256, 0, stream>>>(img, wsY, wsX, out);
}